// GNNModel_12824772345975
// MI455X (gfx1250) — compile-verified
//
#include <hip/hip_runtime.h>

#define N_NODES  50000
#define N_EDGES  800000
#define N_GRAPHS 1024
#define HIDDEN   64
#define MSG_DIM  70          // HIDDEN + EDGE_DIM
#define KPAD     72          // MSG_DIM padded to multiple of 4
#define BN_EPS   1e-5f

typedef __attribute__((ext_vector_type(2))) float v2f;
typedef __attribute__((ext_vector_type(8))) float v8f;

// ---------- order-preserving float <-> uint mapping (for atomic max) ----------
// flip(-inf) = 0x007FFFFF > 0, so a 0-initialized cell means "empty segment".
__device__ __forceinline__ unsigned flipf(float f) {
  unsigned u = __float_as_uint(f);
  return u ^ (unsigned)(((int)u >> 31) | (int)0x80000000);
}
__device__ __forceinline__ float unflipf(unsigned u) {
  unsigned m = ((u >> 31) - 1u) | 0x80000000u;
  return __uint_as_float(u ^ m);
}

// ---------- zero-fill u32 ----------
__global__ void k_init_u32(unsigned* __restrict__ p, int n) {
  int t = blockIdx.x * blockDim.x + threadIdx.x;
  if (t < n) p[t] = 0u;
}

// ---------- edge scatter-max: one wave(32) per edge ----------
__global__ void k_scatter(const float* __restrict__ h,
                          const float* __restrict__ eattr,
                          const int* __restrict__ src,
                          const int* __restrict__ dst,
                          unsigned* __restrict__ agg) {
  int tid  = blockIdx.x * blockDim.x + threadIdx.x;
  int e    = tid >> 5;
  int lane = tid & 31;
  if (e >= N_EDGES) return;
  int s = src[e], d = dst[e];
  const float* hin = h + (size_t)s * HIDDEN;
  unsigned* out    = agg + (size_t)d * KPAD;
  atomicMax(&out[lane],      flipf(hin[lane]));
  atomicMax(&out[lane + 32], flipf(hin[lane + 32]));
  if (lane < 6)
    atomicMax(&out[HIDDEN + lane], flipf(eattr[(size_t)e * 6 + lane]));
}

// ---------- WMMA GEMM: h_pre[N,64] = decode(agg)[N,72] @ Wpad[72,64] + bias ----------
__global__ void __launch_bounds__(256)
k_gemm_wmma(const unsigned* __restrict__ agg,
            const float* __restrict__ W,      // layer slice [70,64] row-major
            const float* __restrict__ bias,   // [64]
            float* __restrict__ hpre) {
  __shared__ float sW[KPAD * HIDDEN];         // 18 KB: K-padded weights
  for (int i = threadIdx.x; i < KPAD * HIDDEN; i += blockDim.x)
    sW[i] = (i < MSG_DIM * HIDDEN) ? W[i] : 0.0f;
  __syncthreads();

  int wave = (blockIdx.x * blockDim.x + threadIdx.x) >> 5;   // 16-row tile id
  int lane = threadIdx.x & 31;
  if (wave < N_NODES / 16) {                 // uniform per wave -> EXEC all ones
    int m0  = wave * 16;
    int l15 = lane & 15;
    int kb  = (lane >> 4) * 2;               // 0 or 2: K sub-pair this half-wave owns
    const unsigned* arow = agg + (size_t)(m0 + l15) * KPAD;

    v8f acc[4] = {v8f{}, v8f{}, v8f{}, v8f{}};
    for (int k0 = 0; k0 < KPAD; k0 += 4) {
      unsigned ua0 = arow[k0 + kb];
      unsigned ua1 = arow[k0 + kb + 1];
      v2f a;
      a.x = ua0 ? unflipf(ua0) : 0.0f;       // empty segment -> 0 (matches seg_max0)
      a.y = ua1 ? unflipf(ua1) : 0.0f;
#pragma unroll
      for (int t = 0; t < 4; ++t) {
        int col = t * 16 + l15;
        v2f bm;
        bm.x = sW[(k0 + kb) * HIDDEN + col];
        bm.y = sW[(k0 + kb + 1) * HIDDEN + col];
        acc[t] = __builtin_amdgcn_wmma_f32_16x16x4_f32(
            /*neg_a=*/false, a, /*neg_b=*/false, bm,
            /*c_mod=*/(short)0, acc[t], /*reuse_a=*/false, /*reuse_b=*/false);
      }
    }

    int mb = (lane >> 4) * 8;                // C/D layout: row half per half-wave
#pragma unroll
    for (int t = 0; t < 4; ++t) {
      int col = t * 16 + l15;
      float bc = bias[col];
#pragma unroll
      for (int r = 0; r < 8; ++r)
        hpre[(size_t)(m0 + mb + r) * HIDDEN + col] = acc[t][r] + bc;
    }
  }
}

// ---------- per-channel sum / sumsq over all nodes ----------
__global__ void k_stats(const float* __restrict__ hpre, float* __restrict__ stats) {
  int col       = threadIdx.x & 63;
  int rowStart  = blockIdx.x * (blockDim.x >> 6) + (threadIdx.x >> 6);
  int rowStride = gridDim.x * (blockDim.x >> 6);
  float s = 0.0f, s2 = 0.0f;
  for (int r = rowStart; r < N_NODES; r += rowStride) {
    float v = hpre[(size_t)r * HIDDEN + col];
    s += v; s2 += v * v;
  }
  atomicAdd(&stats[col], s);
  atomicAdd(&stats[HIDDEN + col], s2);
}

// ---------- BatchNorm + ReLU ----------
__global__ void k_bnrelu(const float* __restrict__ hpre,
                         const float* __restrict__ stats,
                         const float* __restrict__ gamma,
                         const float* __restrict__ beta,
                         float* __restrict__ hout) {
  int t = blockIdx.x * blockDim.x + threadIdx.x;
  if (t >= N_NODES * HIDDEN) return;
  int col = t & 63;
  float inv_n = 1.0f / (float)N_NODES;
  float mean  = stats[col] * inv_n;
  float var   = stats[HIDDEN + col] * inv_n - mean * mean;
  float y = (hpre[t] - mean) * rsqrtf(var + BN_EPS) * gamma[col] + beta[col];
  hout[t] = fmaxf(y, 0.0f);
}

// ---------- graph max-pool (atomic max with flip) ----------
__global__ void k_pool(const float* __restrict__ h,
                       const int* __restrict__ batch,
                       unsigned* __restrict__ pooled) {
  int t = blockIdx.x * blockDim.x + threadIdx.x;
  if (t >= N_NODES * HIDDEN) return;
  int n = t >> 6, c = t & 63;
  int g = batch[n];
  atomicMax(&pooled[(size_t)g * HIDDEN + c], flipf(h[t]));
}

// ---------- final projection [G,64] @ [64,1] + bout ----------
__global__ void k_out(const unsigned* __restrict__ pooled,
                      const float* __restrict__ Wout,
                      const float* __restrict__ bout,
                      float* __restrict__ out) {
  int g = blockIdx.x * blockDim.x + threadIdx.x;
  if (g >= N_GRAPHS) return;
  float acc = 0.0f;
  const unsigned* row = pooled + (size_t)g * HIDDEN;
#pragma unroll
  for (int c = 0; c < HIDDEN; ++c) {
    unsigned u = row[c];
    float v = u ? unflipf(u) : 0.0f;
    acc += v * Wout[c];
  }
  out[g] = acc + bout[0];
}

extern "C" void kernel_launch(void* const* d_in, const int* in_sizes, int n_in,
                              void* d_out, int out_size, void* d_ws, size_t ws_size,
                              hipStream_t stream) {
  const float* x     = (const float*)d_in[0];
  const int*   ei    = (const int*)  d_in[1];
  const float* eattr = (const float*)d_in[2];
  const int*   batch = (const int*)  d_in[3];
  const float* W     = (const float*)d_in[4];
  const float* b     = (const float*)d_in[5];
  const float* gamma = (const float*)d_in[6];
  const float* beta  = (const float*)d_in[7];
  const float* Wout  = (const float*)d_in[8];
  const float* bout  = (const float*)d_in[9];

  char* ws = (char*)d_ws;
  unsigned* agg    = (unsigned*)ws; ws += (size_t)N_NODES * KPAD * 4;
  float*    hcur   = (float*)   ws; ws += (size_t)N_NODES * HIDDEN * 4;
  float*    hpre   = (float*)   ws; ws += (size_t)N_NODES * HIDDEN * 4;
  float*    stats  = (float*)   ws; ws += 128 * 4;
  unsigned* pooled = (unsigned*)ws; ws += (size_t)N_GRAPHS * HIDDEN * 4;

  const int* src = ei;
  const int* dst = ei + N_EDGES;

  const int aggN   = N_NODES * KPAD;
  const int elemN  = N_NODES * HIDDEN;
  const int tilesM = N_NODES / 16;                 // 3125 exactly
  const int gemmBlocks = (tilesM + 7) / 8;         // 8 waves of 32 per block

  for (int layer = 0; layer < 3; ++layer) {
    const float* hin  = (layer == 0) ? x : hcur;
    const float* Wl   = W     + (size_t)layer * MSG_DIM * HIDDEN;
    const float* bl   = b     + (size_t)layer * HIDDEN;
    const float* gl   = gamma + (size_t)layer * HIDDEN;
    const float* btl  = beta  + (size_t)layer * HIDDEN;

    k_init_u32 <<<(aggN + 255) / 256, 256, 0, stream>>>(agg, aggN);
    k_scatter  <<<(N_EDGES * 32) / 256, 256, 0, stream>>>(hin, eattr, src, dst, agg);
    k_gemm_wmma<<<gemmBlocks, 256, 0, stream>>>(agg, Wl, bl, hpre);
    k_init_u32 <<<1, 128, 0, stream>>>((unsigned*)stats, 128);
    k_stats    <<<256, 256, 0, stream>>>(hpre, stats);
    k_bnrelu   <<<(elemN + 255) / 256, 256, 0, stream>>>(hpre, stats, gl, btl, hcur);
  }

  const int poolN = N_GRAPHS * HIDDEN;
  k_init_u32<<<(poolN + 255) / 256, 256, 0, stream>>>(pooled, poolN);
  k_pool    <<<(elemN + 255) / 256, 256, 0, stream>>>(hcur, batch, pooled);
  k_out     <<<(N_GRAPHS + 255) / 256, 256, 0, stream>>>(pooled, Wout, bout, (float*)d_out);
}